// AGNNConv_57767310131236
// MI455X (gfx1250) — compile-verified
//
#include <hip/hip_runtime.h>

#define D 64  // D_IN == D_OUT == 64

typedef __attribute__((ext_vector_type(2))) float v2f;
typedef __attribute__((ext_vector_type(8))) float v8f;

__device__ __forceinline__ void atomAddF32(float* p, float v) {
  // HW fp32 atomic (global_atomic_add_f32), no CAS loop
  unsafeAtomicAdd(p, v);
}

// ---------------------------------------------------------------------------
// Kernel 0: zero the output accumulator (harness poisons d_out with 0xAA)
// ---------------------------------------------------------------------------
__global__ void zero_f32(float* __restrict__ p, int n) {
  int i = blockIdx.x * blockDim.x + threadIdx.x;
  if (i < n) p[i] = 0.0f;
}

// ---------------------------------------------------------------------------
// Kernel 1: Xp = X @ W via V_WMMA_F32_16X16X4_F32
//   One wave per 16x16 output tile. Block = 128 threads = 4 waves covering
//   the 4 N-tiles (D_OUT=64) of one 16-row M tile. K loop: 64/4 = 16 WMMAs.
//
//   A fragment (16x4 f32, 2 VGPRs):  lanes 0-15 hold M=lane, K={0,1};
//                                    lanes 16-31 hold M=lane-16, K={2,3}.
//   B fragment (4x16 f32, 2 VGPRs):  mirrored with N across lanes.
//   C/D (16x16 f32, 8 VGPRs): VGPR j -> row m0 + j + 8*(lane>=16), col = lane&15.
// ---------------------------------------------------------------------------
__global__ void gemm_xw_wmma(const float* __restrict__ X,
                             const float* __restrict__ W,
                             float* __restrict__ Xp, int N) {
  const int lane = threadIdx.x & 31;
  const int ntile = threadIdx.x >> 5;      // 0..3 : which 16-col tile of W
  const int m0 = blockIdx.x * 16;
  const int n0 = ntile * 16;
  const int l15 = lane & 15;
  const int hi  = lane >> 4;               // 0 or 1 (half of wave)
  const int kh  = hi * 2;                  // K offset within the group of 4

  int mload = m0 + l15;
  if (mload >= N) mload = N - 1;           // clamp; garbage rows never stored
  const float* __restrict__ xrow = X + (long)mload * D;

  v8f c = {};
#pragma unroll
  for (int k = 0; k < D; k += 4) {
    v2f a, b;
    a.x = xrow[k + kh + 0];                // VGPR0: K = kh
    a.y = xrow[k + kh + 1];                // VGPR1: K = kh+1
    b.x = W[(k + kh + 0) * D + n0 + l15];  // B row K, striped over lanes by N
    b.y = W[(k + kh + 1) * D + n0 + l15];
    // 8 args: (neg_a, A, neg_b, B, c_mod, C, reuse_a, reuse_b)
    c = __builtin_amdgcn_wmma_f32_16x16x4_f32(false, a, false, b,
                                              (short)0, c, false, false);
  }

  const int mbase = m0 + hi * 8;
  const int n = n0 + l15;
#pragma unroll
  for (int j = 0; j < 8; ++j) {
    int m = mbase + j;
    if (m < N) Xp[(long)m * D + n] = c[j];
  }
}

// ---------------------------------------------------------------------------
// Kernel 2: fused SDDMM + scaled SpMM over the edge list.
//   One wave32 per chunk of EDGES_PER_WAVE consecutive edges; each lane owns
//   2 of the 64 feature dims. Per edge: gather Xp[row] & Xp[col] (L2-resident,
//   12.8 MB << 192 MB), wave-reduce the dot product with shfl_xor, accumulate
//   s * Xp[col] in registers. Because `row` is sorted, we flush the register
//   accumulator with fp32 HW atomics only when the destination row changes —
//   ~16x fewer atomics than per-edge scatter.
// ---------------------------------------------------------------------------
#define EDGES_PER_WAVE 32

__global__ void sddmm_spmm_edges(const float* __restrict__ Xp,
                                 const float* __restrict__ attn_w,
                                 const int* __restrict__ row,
                                 const int* __restrict__ col,
                                 float* __restrict__ out, int E) {
  const int lane = threadIdx.x & 31;
  const long wid = (long)blockIdx.x * (blockDim.x >> 5) + (threadIdx.x >> 5);
  long base = wid * EDGES_PER_WAVE;
  if (base >= E) return;
  long end = base + EDGES_PER_WAVE;
  if (end > E) end = E;

  const float attn = attn_w[0];
  const int d2 = lane * 2;

  int cur = row[base];          // uniform across wave
  float ax = 0.0f, ay = 0.0f;   // per-lane accumulator: dims d2, d2+1

  for (long e = base; e < end; ++e) {
    const int r  = row[e];      // uniform
    const int cc = col[e];      // uniform
    if (r != cur) {             // uniform branch: flush finished segment
      atomAddF32(&out[(long)cur * D + d2 + 0], ax);
      atomAddF32(&out[(long)cur * D + d2 + 1], ay);
      ax = ay = 0.0f;
      cur = r;
    }
    v2f xr = *(const v2f*)(Xp + (long)r  * D + d2);
    v2f xc = *(const v2f*)(Xp + (long)cc * D + d2);
    float p = xr.x * xc.x + xr.y * xc.y;
    p += __shfl_xor(p, 16);
    p += __shfl_xor(p, 8);
    p += __shfl_xor(p, 4);
    p += __shfl_xor(p, 2);
    p += __shfl_xor(p, 1);
    const float s = p * attn;   // edge attention * dot
    ax += s * xc.x;
    ay += s * xc.y;
  }
  atomAddF32(&out[(long)cur * D + d2 + 0], ax);
  atomAddF32(&out[(long)cur * D + d2 + 1], ay);
}

// ---------------------------------------------------------------------------
// Launch
//   d_in[0] = X           [N, 64]  f32
//   d_in[1] = weights     [64, 64] f32
//   d_in[2] = attention_w [1]      f32
//   d_in[3] = row         [E]      i32 (sorted)
//   d_in[4] = col         [E]      i32
//   d_out   = out         [N, 64]  f32
//   d_ws    = Xp          [N, 64]  f32 (12.8 MB)
// ---------------------------------------------------------------------------
extern "C" void kernel_launch(void* const* d_in, const int* in_sizes, int n_in,
                              void* d_out, int out_size, void* d_ws, size_t ws_size,
                              hipStream_t stream) {
  const float* X    = (const float*)d_in[0];
  const float* W    = (const float*)d_in[1];
  const float* attn = (const float*)d_in[2];
  const int*   row  = (const int*)d_in[3];
  const int*   col  = (const int*)d_in[4];
  float* out = (float*)d_out;
  float* Xp  = (float*)d_ws;

  const int N = in_sizes[0] / D;   // 50000
  const int E = in_sizes[3];       // 800000

  // 1) zero output
  {
    int n = N * D;
    zero_f32<<<(n + 255) / 256, 256, 0, stream>>>(out, n);
  }
  // 2) dense GEMM (WMMA f32 16x16x4): one block per 16-row tile, 4 waves cover N
  {
    int mtiles = (N + 15) / 16;    // 3125
    gemm_xw_wmma<<<mtiles, 128, 0, stream>>>(X, W, Xp, N);
  }
  // 3) fused SDDMM + SpMM over edges
  {
    long waves  = ((long)E + EDGES_PER_WAVE - 1) / EDGES_PER_WAVE;  // 25000
    long blocks = (waves + 7) / 8;                                  // 8 waves/block
    sddmm_spmm_edges<<<(int)blocks, 256, 0, stream>>>(Xp, attn, row, col, out, E);
  }
}